// CBAM_68255620268841
// MI455X (gfx1250) — compile-verified
//
#include <hip/hip_runtime.h>
#include <hip/hip_bf16.h>
#include <math.h>

// CBAM: x[16,256,128,128] f32. Memory-bound: 3 reads of x + 1 write ~ 1GB
// -> ~44us at 23.3 TB/s. Tiny MLP done with V_WMMA_F32_16X16X4_F32 (fp32 WMMA).

#define B_  16
#define C_  256
#define CR_ 16
#define H_  128
#define W_  128
#define HW_ 16384          // 128*128
#define KS_ 7

typedef __attribute__((ext_vector_type(2))) float v2f;
typedef __attribute__((ext_vector_type(8))) float v8f;

// ---------------------------------------------------------------------------
// Kernel 1: per-(b,c) mean & max over HxW. One block per (b,c), 256 threads,
// float4 (global_load_b128) streaming + shared-memory tree reduction.
// ---------------------------------------------------------------------------
__global__ __launch_bounds__(256)
void k_channel_stats(const float4* __restrict__ x4,
                     float* __restrict__ avg, float* __restrict__ mx)
{
    const int bc  = blockIdx.x;           // 0 .. B*C-1
    const int tid = threadIdx.x;
    const float4* p = x4 + (size_t)bc * (HW_ / 4);
    float s = 0.f, m = -3.402823466e38f;
#pragma unroll 4
    for (int i = tid; i < HW_ / 4; i += 256) {
        float4 v = p[i];
        s += v.x + v.y + v.z + v.w;
        m = fmaxf(m, fmaxf(fmaxf(v.x, v.y), fmaxf(v.z, v.w)));
    }
    __shared__ float ss[256];
    __shared__ float sm[256];
    ss[tid] = s; sm[tid] = m;
    __syncthreads();
    for (int st = 128; st > 0; st >>= 1) {
        if (tid < st) {
            ss[tid] += ss[tid + st];
            sm[tid] = fmaxf(sm[tid], sm[tid + st]);
        }
        __syncthreads();
    }
    if (tid == 0) {
        avg[bc] = ss[0] * (1.f / (float)HW_);
        mx[bc]  = sm[0];
    }
}

// ---------------------------------------------------------------------------
// Kernel 2: channel-attention MLP on one wave32 using fp32 WMMA.
//   h   = relu(avg @ w1^T) + relu(mx @ w1^T)        [16 x 16]
//   att = sigmoid(h @ w2^T)                         [16 x 256]
// A 16x4 f32 layout: lane m=lane&15, vgpr j holds A[m][koff+j], koff=(lane>=16)?2:0
// B 4x16  f32 layout: mirrored, B[koff+j][lane&15]
// C/D 16x16 f32: d[r] = D[r + (lane>=16?8:0)][lane&15]
// ---------------------------------------------------------------------------
__global__ __launch_bounds__(32)
void k_mlp_wmma(const float* __restrict__ avg, const float* __restrict__ mx,
                const float* __restrict__ w1,  const float* __restrict__ w2,
                float* __restrict__ ch_att)
{
    const int lane = threadIdx.x;
    const int m    = lane & 15;            // A row / B column index
    const int koff = (lane >> 4) << 1;     // 0 or 2
    const int rb   = (lane >> 4) << 3;     // C/D row base: 0 or 8

    v8f acc_a = {};
    v8f acc_m = {};
    // GEMM1: M=16 (batch), N=16 (hidden), K=256 (channels)
    for (int kb = 0; kb < C_; kb += 4) {
        v2f a1, a2, bb;
        a1.x = avg[m * C_ + kb + koff];
        a1.y = avg[m * C_ + kb + koff + 1];
        a2.x = mx [m * C_ + kb + koff];
        a2.y = mx [m * C_ + kb + koff + 1];
        bb.x = w1 [m * C_ + kb + koff];      // B[k][n] = w1[n][k], n == lane&15
        bb.y = w1 [m * C_ + kb + koff + 1];
        acc_a = __builtin_amdgcn_wmma_f32_16x16x4_f32(
            false, a1, false, bb, (short)0, acc_a, false, false);
        acc_m = __builtin_amdgcn_wmma_f32_16x16x4_f32(
            false, a2, false, bb, (short)0, acc_m, false, false);
    }

    // h = relu(acc_a) + relu(acc_m), then transpose D-layout -> A-layout via LDS
    __shared__ float hlds[16 * 16];
#pragma unroll
    for (int r = 0; r < 8; ++r) {
        float hv = fmaxf(acc_a[r], 0.f) + fmaxf(acc_m[r], 0.f);
        hlds[(rb + r) * 16 + m] = hv;
    }
    __syncthreads();

    // GEMM2: M=16 (batch), N=256 (channels, 16 tiles), K=16 (hidden)
    for (int t = 0; t < 16; ++t) {
        v8f acc = {};
#pragma unroll
        for (int kb = 0; kb < CR_; kb += 4) {
            v2f a, bb;
            a.x  = hlds[m * 16 + kb + koff];
            a.y  = hlds[m * 16 + kb + koff + 1];
            bb.x = w2[(t * 16 + m) * CR_ + kb + koff];   // B[k][n] = w2[n_glob][k]
            bb.y = w2[(t * 16 + m) * CR_ + kb + koff + 1];
            acc = __builtin_amdgcn_wmma_f32_16x16x4_f32(
                false, a, false, bb, (short)0, acc, false, false);
        }
#pragma unroll
        for (int r = 0; r < 8; ++r) {
            float v = acc[r];
            ch_att[(rb + r) * C_ + t * 16 + m] = 1.f / (1.f + __expf(-v));
        }
    }
}

// ---------------------------------------------------------------------------
// Kernel 3: spatial mean/max over channels of (x * ch_att). One thread/pixel.
// Writes sp[b][0][h][w]=mean, sp[b][1][h][w]=max.
// ---------------------------------------------------------------------------
__global__ __launch_bounds__(256)
void k_spatial_stats(const float* __restrict__ x, const float* __restrict__ ch_att,
                     float* __restrict__ sp)
{
    const int p  = blockIdx.x * 256 + threadIdx.x;   // 0 .. B*HW-1
    const int b  = p >> 14;
    const int hw = p & (HW_ - 1);
    const float* xb = x + (size_t)b * C_ * HW_;
    const float* ca = ch_att + b * C_;
    float s = 0.f, m = -3.402823466e38f;
#pragma unroll 4
    for (int c = 0; c < C_; ++c) {
        float v = xb[(size_t)c * HW_ + hw] * ca[c];
        s += v;
        m = fmaxf(m, v);
    }
    sp[((size_t)(b * 2)     ) * HW_ + hw] = s * (1.f / (float)C_);
    sp[((size_t)(b * 2 + 1) ) * HW_ + hw] = m;
}

// ---------------------------------------------------------------------------
// Kernel 4: 7x7 conv (2 in-ch -> 1) + sigmoid on the 2MB stats plane.
// ---------------------------------------------------------------------------
__global__ __launch_bounds__(256)
void k_conv_sp(const float* __restrict__ sp, const float* __restrict__ wconv,
               float* __restrict__ sp_att)
{
    const int p  = blockIdx.x * 256 + threadIdx.x;   // 0 .. B*HW-1
    const int b  = p >> 14;
    const int hw = p & (HW_ - 1);
    const int h  = hw >> 7;
    const int w  = hw & (W_ - 1);
    float acc = 0.f;
#pragma unroll
    for (int ci = 0; ci < 2; ++ci) {
        const float* plane = sp + (size_t)(b * 2 + ci) * HW_;
#pragma unroll
        for (int ky = 0; ky < KS_; ++ky) {
            const int iy = h + ky - KS_ / 2;
            if (iy < 0 || iy >= H_) continue;
#pragma unroll
            for (int kx = 0; kx < KS_; ++kx) {
                const int ix = w + kx - KS_ / 2;
                if (ix < 0 || ix >= W_) continue;
                acc += plane[iy * W_ + ix] * wconv[(ci * KS_ + ky) * KS_ + kx];
            }
        }
    }
    sp_att[p] = 1.f / (1.f + __expf(-acc));
}

// ---------------------------------------------------------------------------
// Kernel 5: out = x * ch_att[b,c] * sp_att[b,h,w], float4 streaming.
// ---------------------------------------------------------------------------
__global__ __launch_bounds__(256)
void k_apply(const float4* __restrict__ x4, const float* __restrict__ ch_att,
             const float4* __restrict__ sp_att4, float4* __restrict__ out4)
{
    const int p4 = blockIdx.x * 256 + threadIdx.x;   // 0 .. B*C*HW/4 - 1
    const int bc  = p4 >> 12;                        // 4096 float4 per (b,c) plane
    const int b   = bc >> 8;
    const int hw4 = p4 & 4095;
    const float ca = ch_att[bc];
    float4 v  = x4[p4];
    float4 sa = sp_att4[(b << 12) + hw4];
    v.x *= ca * sa.x;
    v.y *= ca * sa.y;
    v.z *= ca * sa.z;
    v.w *= ca * sa.w;
    out4[p4] = v;
}

// ---------------------------------------------------------------------------
extern "C" void kernel_launch(void* const* d_in, const int* in_sizes, int n_in,
                              void* d_out, int out_size, void* d_ws, size_t ws_size,
                              hipStream_t stream) {
    const float* x     = (const float*)d_in[0];   // [16,256,128,128]
    const float* w1    = (const float*)d_in[1];   // [16,256]
    const float* w2    = (const float*)d_in[2];   // [256,16]
    const float* wconv = (const float*)d_in[3];   // [1,2,7,7]
    float* out = (float*)d_out;

    // Workspace layout (floats): avg | mx | ch_att | sp[B,2,H,W] | sp_att[B,H,W]
    float* ws     = (float*)d_ws;
    float* avg    = ws;                                   // 4096
    float* mx     = ws + B_ * C_;                         // 4096
    float* ch_att = ws + 2 * B_ * C_;                     // 4096
    float* sp     = ws + 3 * B_ * C_;                     // 16*2*16384 = 524288
    float* sp_att = sp + B_ * 2 * HW_;                    // 262144

    // 1) per-(b,c) mean/max of x
    k_channel_stats<<<B_ * C_, 256, 0, stream>>>((const float4*)x, avg, mx);
    // 2) fp32-WMMA MLP -> ch_att  (single wave32)
    k_mlp_wmma<<<1, 32, 0, stream>>>(avg, mx, w1, w2, ch_att);
    // 3) per-pixel channel mean/max of x*ch_att
    k_spatial_stats<<<B_ * HW_ / 256, 256, 0, stream>>>(x, ch_att, sp);
    // 4) 7x7 conv + sigmoid -> sp_att
    k_conv_sp<<<B_ * HW_ / 256, 256, 0, stream>>>(sp, wconv, sp_att);
    // 5) out = x * ch_att * sp_att
    k_apply<<<(B_ * C_ * HW_ / 4) / 256, 256, 0, stream>>>(
        (const float4*)x, ch_att, (const float4*)sp_att, (float4*)out);
}